// PositionWiseSpatialAttention_82566451299288
// MI455X (gfx1250) — compile-verified
//
#include <hip/hip_runtime.h>

typedef __attribute__((ext_vector_type(2))) float v2f;
typedef __attribute__((ext_vector_type(8))) float v8f;

#define BB 8
#define NN 1024
#define TT 12
#define DD 64
#define OO 64
#define PAD 68            // floats per padded LDS row (16B aligned, bank-conflict free)
#define KTILE 64          // columns (m) per LDS tile
#define NTILES (NN / KTILE)
#define ROWS_PER_WG 128
#define WAVES 8

// Async copy of a 64x64 f32 tile (row stride `rstride` floats) into a padded
// LDS tile using the CDNA5 direct global->LDS path (ASYNCcnt-tracked).
// 256 threads: 4 threads per row, each issuing 4x16B async transfers.
__device__ __forceinline__ void coop_load_async_64x64(float* lds, const float* src,
                                                      int rstride, int tid) {
  const int r  = tid >> 2;
  const int cb = (tid & 3) * 16;
  const float* p = src + (size_t)r * rstride + cb;
  const unsigned lds0 = (unsigned)(size_t)(lds + r * PAD + cb);  // LDS byte addr = low 32 bits
#pragma unroll
  for (int i = 0; i < 4; ++i) {
    const unsigned long long ga = (unsigned long long)(size_t)(p + 4 * i);
    const unsigned lo = lds0 + 16u * i;
    asm volatile("global_load_async_to_lds_b128 %0, %1, off"
                 :: "v"(lo), "v"(ga) : "memory");
  }
}

__device__ __forceinline__ void wait_async0() {
  asm volatile("s_wait_asynccnt 0x0" ::: "memory");
}

__launch_bounds__(256, 1)
__global__ void psa_fused_kernel(const float* __restrict__ x,
                                 const float* __restrict__ adj,
                                 const float* __restrict__ w,
                                 float* __restrict__ out) {
  __shared__ float kbuf[2][KTILE * PAD];        // double-buffered K/V (later W) tile
  __shared__ float stage[WAVES * 16 * PAD];     // per-wave C->A relayout staging

  const int tid  = threadIdx.x;
  const int lane = tid & 31;
  const int wave = tid >> 5;
  const int lr   = lane & 15;   // column / M index within 16
  const int hh   = lane >> 4;   // half-wave select
  const int g    = blockIdx.y;  // 0..95
  const int b    = g / TT;
  const int t    = g % TT;
  const int rowbase = blockIdx.x * ROWS_PER_WG + wave * 16;
  float* pst = &stage[wave * 16 * PAD];

  const v8f zero8 = {0.f, 0.f, 0.f, 0.f, 0.f, 0.f, 0.f, 0.f};

  // ---- Q tile (16 rows x 64) as 16 A-fragments (K=4 each), kept in registers.
  v2f q[16];
  {
    const int row = rowbase + lr;
    const float* xp = x + (((size_t)b * NN + row) * TT + t) * DD;
#pragma unroll
    for (int kc = 0; kc < 16; ++kc) {
      const float* p = xp + 4 * kc + 2 * hh;
      v2f a = {p[0], p[1]};
      q[kc] = a;
    }
  }

  // Per-lane softmax stats: VGPR j of a C fragment <-> row (j + 8*hh).
  float m[8], lsum[8];
#pragma unroll
  for (int j = 0; j < 8; ++j) { m[j] = -1e30f; lsum[j] = 0.f; }

  const float* xg_base = x + (((size_t)b * NN) * TT + t) * DD; // row stride TT*DD

  // ================= Sweep 1: online row max / sum-exp =================
  coop_load_async_64x64(kbuf[0], xg_base, TT * DD, tid);
  wait_async0();
  __syncthreads();

  for (int mt = 0; mt < NTILES; ++mt) {
    const float* cur = kbuf[mt & 1];
    if (mt + 1 < NTILES)
      coop_load_async_64x64(kbuf[(mt + 1) & 1],
                            xg_base + (size_t)((mt + 1) * KTILE) * (TT * DD),
                            TT * DD, tid);

    v8f s[4];
#pragma unroll
    for (int nt = 0; nt < 4; ++nt) {
      v8f acc = zero8;
#pragma unroll
      for (int kc = 0; kc < 16; ++kc) {
        // B = Ktile^T : B[k][n] = Ktile[n][k] -> float2 from LDS row (nt*16+lr)
        const float* p = &cur[(nt * 16 + lr) * PAD + 4 * kc + 2 * hh];
        v2f bf = {p[0], p[1]};
        acc = __builtin_amdgcn_wmma_f32_16x16x4_f32(false, q[kc], false, bf,
                                                    (short)0, acc, false, false);
      }
#pragma unroll
      for (int j = 0; j < 8; ++j) acc[j] *= 0.125f;  // 1/sqrt(64)
      s[nt] = acc;
    }

#pragma unroll
    for (int j = 0; j < 8; ++j) {
      float tm = fmaxf(fmaxf(s[0][j], s[1][j]), fmaxf(s[2][j], s[3][j]));
#pragma unroll
      for (int xm = 8; xm >= 1; xm >>= 1) tm = fmaxf(tm, __shfl_xor(tm, xm, 32));
      const float mn   = fmaxf(m[j], tm);
      const float corr = __expf(m[j] - mn);
      float e = __expf(s[0][j] - mn) + __expf(s[1][j] - mn) +
                __expf(s[2][j] - mn) + __expf(s[3][j] - mn);
#pragma unroll
      for (int xm = 8; xm >= 1; xm >>= 1) e += __shfl_xor(e, xm, 32);
      lsum[j] = lsum[j] * corr + e;
      m[j] = mn;
    }

    wait_async0();       // our async stores into the next buffer have landed
    __syncthreads();     // everyone done reading cur / loading next
  }

  float invl[8];
#pragma unroll
  for (int j = 0; j < 8; ++j) invl[j] = 1.0f / lsum[j];

  // ================= Sweep 2: agg = (adj .* softmax) @ V =================
  v8f agg[4] = {zero8, zero8, zero8, zero8};

  coop_load_async_64x64(kbuf[0], xg_base, TT * DD, tid);
  wait_async0();
  __syncthreads();

  for (int mt = 0; mt < NTILES; ++mt) {
    const float* cur = kbuf[mt & 1];
    if (mt + 1 < NTILES)
      coop_load_async_64x64(kbuf[(mt + 1) & 1],
                            xg_base + (size_t)((mt + 1) * KTILE) * (TT * DD),
                            TT * DD, tid);

    // Recompute scores, convert to masked probabilities, stage row-major.
#pragma unroll
    for (int nt = 0; nt < 4; ++nt) {
      v8f acc = zero8;
#pragma unroll
      for (int kc = 0; kc < 16; ++kc) {
        const float* p = &cur[(nt * 16 + lr) * PAD + 4 * kc + 2 * hh];
        v2f bf = {p[0], p[1]};
        acc = __builtin_amdgcn_wmma_f32_16x16x4_f32(false, q[kc], false, bf,
                                                    (short)0, acc, false, false);
      }
#pragma unroll
      for (int j = 0; j < 8; ++j) {
        const int row = rowbase + j + 8 * hh;
        const int col = mt * KTILE + nt * 16 + lr;
        const float a  = adj[(size_t)row * NN + col];
        const float pv = a * __expf(acc[j] * 0.125f - m[j]) * invl[j];
        pst[(j + 8 * hh) * PAD + nt * 16 + lr] = pv;
      }
    }
    asm volatile("s_wait_dscnt 0x0" ::: "memory");  // same-wave cross-lane LDS RAW

    // agg += P(16x64) @ V(64x64), V = cur tile.
#pragma unroll
    for (int dt = 0; dt < 4; ++dt) {
      v8f acc = agg[dt];
#pragma unroll
      for (int kc = 0; kc < 16; ++kc) {
        const float* pa = &pst[lr * PAD + 4 * kc + 2 * hh];
        v2f af = {pa[0], pa[1]};
        const int kk = 4 * kc + 2 * hh;
        v2f bf = {cur[kk * PAD + dt * 16 + lr],
                  cur[(kk + 1) * PAD + dt * 16 + lr]};
        acc = __builtin_amdgcn_wmma_f32_16x16x4_f32(false, af, false, bf,
                                                    (short)0, acc, false, false);
      }
      agg[dt] = acc;
    }

    wait_async0();
    __syncthreads();
  }

  // ================= Projection: out = relu(agg @ W), transposed store ===
  coop_load_async_64x64(kbuf[0], w, OO, tid);   // reuse tile buffer for W (64x64)
  wait_async0();
  __syncthreads();
  const float* wl = kbuf[0];

#pragma unroll
  for (int dt = 0; dt < 4; ++dt)
#pragma unroll
    for (int j = 0; j < 8; ++j)
      pst[(j + 8 * hh) * PAD + dt * 16 + lr] = agg[dt][j];
  asm volatile("s_wait_dscnt 0x0" ::: "memory");

#pragma unroll
  for (int ot = 0; ot < 4; ++ot) {
    v8f acc = zero8;
#pragma unroll
    for (int kc = 0; kc < 16; ++kc) {
      const float* pa = &pst[lr * PAD + 4 * kc + 2 * hh];
      v2f af = {pa[0], pa[1]};
      const int kk = 4 * kc + 2 * hh;
      v2f bf = {wl[kk * PAD + ot * 16 + lr],
                wl[(kk + 1) * PAD + ot * 16 + lr]};
      acc = __builtin_amdgcn_wmma_f32_16x16x4_f32(false, af, false, bf,
                                                  (short)0, acc, false, false);
    }
#pragma unroll
    for (int j = 0; j < 8; ++j) {
      const int row = rowbase + j + 8 * hh;
      const int o   = ot * 16 + lr;
      out[(((size_t)b * NN + row) * TT + t) * OO + o] = fmaxf(acc[j], 0.f);
    }
  }
}

extern "C" void kernel_launch(void* const* d_in, const int* in_sizes, int n_in,
                              void* d_out, int out_size, void* d_ws, size_t ws_size,
                              hipStream_t stream) {
  (void)in_sizes; (void)n_in; (void)out_size; (void)d_ws; (void)ws_size;
  const float* x   = (const float*)d_in[0];
  const float* adj = (const float*)d_in[1];
  const float* w   = (const float*)d_in[2];
  float* out = (float*)d_out;

  dim3 grid(NN / ROWS_PER_WG, BB * TT);  // (8, 96) workgroups
  psa_fused_kernel<<<grid, 256, 0, stream>>>(x, adj, w, out);
}